// smoothTransformer3D_83614423318531
// MI455X (gfx1250) — compile-verified
//
#include <hip/hip_runtime.h>

// MI455X (gfx1250) smooth-transformer-3D.
// Memory-bound pipeline (~600MB traffic -> ~25us @ 23.3 TB/s).
// Z-axis cumsum implemented as blocked triangular matmul on the WMMA path
// (V_WMMA_F32_16X16X4_F32 chained x4 => 16x16 inclusive scan per block).

typedef __attribute__((ext_vector_type(2))) float v2f;
typedef __attribute__((ext_vector_type(8))) float v8f;

#define DIMV 128
#define NVOX (4l * 128l * 128l * 128l)   // 8,388,608
#define EPSV 1e-7f

// ---------------------------------------------------------------------------
// K1: dg = 2/(1+exp(-x)) and deinterleave 3 channels into planar ws arrays.
// Coalesced 12B/thread reads, planar writes stay resident in 192MB L2.
// ---------------------------------------------------------------------------
__global__ void st3d_logistic_deint(const float* __restrict__ defgrad,
                                    float* __restrict__ ws, long n) {
  long i = (long)blockIdx.x * blockDim.x + threadIdx.x;
  if (i >= n) return;
  float a = defgrad[3 * i + 0];
  float b = defgrad[3 * i + 1];
  float c = defgrad[3 * i + 2];
  ws[i]         = 2.0f / (1.0f + __expf(-a));
  ws[n + i]     = 2.0f / (1.0f + __expf(-b));
  ws[2 * n + i] = 2.0f / (1.0f + __expf(-c));
}

// ---------------------------------------------------------------------------
// K2: scan along X (stride 128*128 within planar channel 0).
// One thread per (b,y,z) line; adjacent lanes -> adjacent z -> coalesced.
// Pass1: inclusive cumsum, write sampling_grid (cs-1) in place.
// Pass2: reload (L2-hot), emit grid_norm / grid_inverse channel 0.
// ---------------------------------------------------------------------------
__global__ void st3d_scan_x(float* __restrict__ sg, float* __restrict__ out, long Nv) {
  int t = blockIdx.x * blockDim.x + threadIdx.x;       // 4*128*128 threads
  int b = t >> 14;
  int yz = t & 16383;
  float* p = sg + ((long)b << 21) + yz;
  const long stride = DIMV * DIMV;
  float cs = 0.f, cs0 = 0.f;
  for (int i = 0; i < DIMV; ++i) {
    float v = p[(long)i * stride];
    cs += v;
    if (i == 0) cs0 = cs;
    p[(long)i * stride] = cs - 1.0f;
  }
  float rden = 1.0f / (cs - cs0 + EPSV);
  float* gn = out + 2 * Nv;
  float* gi = out + 5 * Nv;
  for (int i = 0; i < DIMV; ++i) {
    float g = p[(long)i * stride] + 1.0f;              // cs_i
    float nrm = 127.0f * (g - cs0) * rden;
    long v = ((long)b << 21) + (long)i * stride + yz;
    gn[3 * v + 0] = nrm;
    gi[3 * v + 0] = 2.0f * (float)i - nrm;
  }
}

// ---------------------------------------------------------------------------
// K3: scan along Y (stride 128 within planar channel 1).
// Thread per (b,x,z); adjacent lanes -> adjacent z -> coalesced.
// ---------------------------------------------------------------------------
__global__ void st3d_scan_y(float* __restrict__ sg, float* __restrict__ out, long Nv) {
  int t = blockIdx.x * blockDim.x + threadIdx.x;       // 4*128*128 threads
  int z = t & 127;
  int bx = t >> 7;                                     // b*128 + x
  float* p = sg + ((long)bx << 14) + z;
  const long stride = DIMV;
  float cs = 0.f, cs0 = 0.f;
  for (int i = 0; i < DIMV; ++i) {
    float v = p[(long)i * stride];
    cs += v;
    if (i == 0) cs0 = cs;
    p[(long)i * stride] = cs - 1.0f;
  }
  float rden = 1.0f / (cs - cs0 + EPSV);
  float* gn = out + 2 * Nv;
  float* gi = out + 5 * Nv;
  for (int i = 0; i < DIMV; ++i) {
    float g = p[(long)i * stride] + 1.0f;
    float nrm = 127.0f * (g - cs0) * rden;
    long v = ((long)bx << 14) + (long)i * stride + z;
    gn[3 * v + 1] = nrm;
    gi[3 * v + 1] = 2.0f * (float)i - nrm;
  }
}

// ---------------------------------------------------------------------------
// K4: scan along Z (contiguous 128-float lines) via WMMA.
// Wave handles 16 lines (a 16x128 tile = eight 16x16 blocks).
// Within-block inclusive scan: S = A * U, U upper-triangular ones,
// realized as 4 chained v_wmma_f32_16x16x4_f32 (K=4 each).
// Inter-block carry: shuffle column 15 of D (lanes 15/31) and accumulate.
// A layout (ISA 7.12.2): lanes 0-15 rows M, K=0..1; lanes 16-31 rows M, K=2..3.
// D layout: lane<16 -> N=lane, M=vgpr; lane>=16 -> N=lane-16, M=8+vgpr.
// ---------------------------------------------------------------------------
__global__ void st3d_scan_z_wmma(float* __restrict__ sgz, float* __restrict__ out, long Nv) {
  const int lane = threadIdx.x & 31;
  const int wave = threadIdx.x >> 5;
  const int half = lane >> 4;          // K/M half select
  const int nl   = lane & 15;          // A-row for loads, D-column for results
  const long lineBase = ((long)blockIdx.x * 8 + wave) * 16;
  float* tile = sgz + lineBase * 128;

  // Triangular B operand: U[k][n] = (k <= n), per K-split j, per lane half.
  v2f bm[4];
#pragma unroll
  for (int j = 0; j < 4; ++j) {
    int k = 4 * j + 2 * half;
    bm[j][0] = (k     <= nl) ? 1.0f : 0.0f;
    bm[j][1] = (k + 1 <= nl) ? 1.0f : 0.0f;
  }

  float carry[8], cs0[8];
#pragma unroll
  for (int i = 0; i < 8; ++i) { carry[i] = 0.0f; cs0[i] = 0.0f; }
  const int srcHi = (lane & 16) | 15;  // lane holding column 15 of my half
  const int srcLo = (lane & 16);       // lane holding column 0 of my half
  const int myrow = half * 8;

#pragma unroll
  for (int cb = 0; cb < 8; ++cb) {
    v8f acc = {};
#pragma unroll
    for (int j = 0; j < 4; ++j) {
      const float* ap = tile + (long)nl * 128 + 16 * cb + 4 * j + 2 * half;
      v2f a = *(const v2f*)ap;
      acc = __builtin_amdgcn_wmma_f32_16x16x4_f32(
          false, a, false, bm[j], (short)0, acc, false, false);
    }
    float outv[8];
#pragma unroll
    for (int i = 0; i < 8; ++i) {
      outv[i] = acc[i] + carry[i];                       // add inter-block carry
      carry[i] = __shfl(outv[i], srcHi, 32);             // running total of my row
      if (cb == 0) cs0[i] = __shfl(outv[i], srcLo, 32);  // cs at z=0 for my row
      tile[(long)(myrow + i) * 128 + 16 * cb + nl] = outv[i] - 1.0f;  // sampling_grid
    }
  }

  // Pass 2: emit grid_norm / grid_inverse channel 2 (reload is L0-hot).
  float rden[8];
#pragma unroll
  for (int i = 0; i < 8; ++i) rden[i] = 1.0f / (carry[i] - cs0[i] + EPSV);
  float* gn = out + 2 * Nv;
  float* gi = out + 5 * Nv;
#pragma unroll
  for (int cb = 0; cb < 8; ++cb) {
#pragma unroll
    for (int i = 0; i < 8; ++i) {
      int col = 16 * cb + nl;
      float cs = tile[(long)(myrow + i) * 128 + col] + 1.0f;
      float nrm = 127.0f * (cs - cs0[i]) * rden[i];
      long v = (lineBase + myrow + i) * 128 + col;
      gn[3 * v + 2] = nrm;
      gi[3 * v + 2] = 2.0f * (float)col - nrm;
    }
  }
}

// ---------------------------------------------------------------------------
// Trilinear sampler matching the reference exactly (clamped corners, deltas
// taken against the *clamped* floor coordinate).
// ---------------------------------------------------------------------------
__device__ __forceinline__ int st3d_clamp(int v) {
  return v < 0 ? 0 : (v > 127 ? 127 : v);
}

__device__ __forceinline__ float st3d_trilerp(const float* __restrict__ vol,
                                              float xs, float ys, float zs) {
  int x0 = (int)floorf(xs), y0 = (int)floorf(ys), z0 = (int)floorf(zs);
  int x0c = st3d_clamp(x0), x1c = st3d_clamp(x0 + 1);
  int y0c = st3d_clamp(y0), y1c = st3d_clamp(y0 + 1);
  int z0c = st3d_clamp(z0), z1c = st3d_clamp(z0 + 1);
  float xd = xs - (float)x0c;
  float yd = ys - (float)y0c;
  float zd = zs - (float)z0c;
  int bx0 = x0c << 14, bx1 = x1c << 14;
  int by0 = y0c << 7,  by1 = y1c << 7;
  float Ia = vol[bx0 + by0 + z0c];
  float Ib = vol[bx0 + by0 + z1c];
  float Ic = vol[bx0 + by1 + z0c];
  float Id = vol[bx0 + by1 + z1c];
  float Ie = vol[bx1 + by0 + z0c];
  float If = vol[bx1 + by0 + z1c];
  float Ig = vol[bx1 + by1 + z0c];
  float Ih = vol[bx1 + by1 + z1c];
  float Cae = Ia * (1.0f - xd) + Ie * xd;
  float Cbf = Ib * (1.0f - xd) + If * xd;
  float Ccg = Ic * (1.0f - xd) + Ig * xd;
  float Cdh = Id * (1.0f - xd) + Ih * xd;
  float Caecg = Cae * (1.0f - yd) + Ccg * yd;
  float Cbfdh = Cbf * (1.0f - yd) + Cdh * yd;
  return Caecg * (1.0f - zd) + Cbfdh * zd;
}

// K5: mov_def = resample(mov, sampling_grid)  (planar sg channels from ws)
__global__ void st3d_gather_mov(const float* __restrict__ mov,
                                const float* __restrict__ sgx,
                                const float* __restrict__ sgy,
                                const float* __restrict__ sgz,
                                float* __restrict__ out, long Nv) {
  long i = (long)blockIdx.x * blockDim.x + threadIdx.x;
  if (i >= Nv) return;
  int b = (int)(i >> 21);
  const float* vol = mov + ((long)b << 21);
  out[i] = st3d_trilerp(vol, sgx[i], sgy[i], sgz[i]);
}

// K6: ref_def = resample(ref, grid_inverse)  (grid_inverse read back from d_out)
__global__ void st3d_gather_ref(const float* __restrict__ ref,
                                float* __restrict__ out, long Nv) {
  long i = (long)blockIdx.x * blockDim.x + threadIdx.x;
  if (i >= Nv) return;
  int b = (int)(i >> 21);
  const float* gi = out + 5 * Nv;
  float xs = gi[3 * i + 0];
  float ys = gi[3 * i + 1];
  float zs = gi[3 * i + 2];
  const float* vol = ref + ((long)b << 21);
  out[Nv + i] = st3d_trilerp(vol, xs, ys, zs);
}

extern "C" void kernel_launch(void* const* d_in, const int* in_sizes, int n_in,
                              void* d_out, int out_size, void* d_ws, size_t ws_size,
                              hipStream_t stream) {
  const float* mov = (const float*)d_in[0];
  const float* ref = (const float*)d_in[1];
  const float* dgr = (const float*)d_in[2];
  float* out = (float*)d_out;
  float* ws  = (float*)d_ws;     // 3*NVOX floats: planar [dg->sampling_grid] x3
  const long Nv = NVOX;

  st3d_logistic_deint<<<(int)((Nv + 255) / 256), 256, 0, stream>>>(dgr, ws, Nv);
  st3d_scan_x<<<65536 / 256, 256, 0, stream>>>(ws, out, Nv);
  st3d_scan_y<<<65536 / 256, 256, 0, stream>>>(ws + Nv, out, Nv);
  st3d_scan_z_wmma<<<512, 256, 0, stream>>>(ws + 2 * Nv, out, Nv);
  st3d_gather_mov<<<(int)((Nv + 255) / 256), 256, 0, stream>>>(
      mov, ws, ws + Nv, ws + 2 * Nv, out, Nv);
  st3d_gather_ref<<<(int)((Nv + 255) / 256), 256, 0, stream>>>(ref, out, Nv);
}